// GaussianKernel_27444841021749
// MI455X (gfx1250) — compile-verified
//
#include <hip/hip_runtime.h>
#include <stdint.h>

typedef float v2f __attribute__((ext_vector_type(2)));
typedef float v8f __attribute__((ext_vector_type(8)));

#define KS    15
#define PADR  7
#define TILE  64
#define IN_T  (TILE + KS - 1)   // 78 valid halo rows/cols
#define TROWS 80                // 78 real + 2 zero rows (V reach handled in hbuf)
#define TSTR  84                // stride: 20L mod 64 distinct -> conflict-free column b64 reads
#define HROWS 80                // 78 real + 2 zero (WMMA K=32 over-reach rows, zero weights)
#define HSTR  72                // rows+2 -> +16 banks: conflict-free V-pass B loads
#define IMG   512

__global__ __launch_bounds__(256)
void gauss_blur_fused(const float* __restrict__ x,
                      const float* __restrict__ sigma,
                      float* __restrict__ out,
                      int channels) {
    __shared__ float tile[TROWS * TSTR];   // 26.9 KB input halo tile
    __shared__ float hbuf[HROWS * HSTR];   // 23.0 KB horizontally blurred tile
    __shared__ float hwpad[64];            // raw g at [16..30], zeros elsewhere
    __shared__ float vwpad[64];            // g/( (Σg)^2 + eps ) at [16..30]

    const int tid = threadIdx.x;
    const int bc  = blockIdx.z;
    const int b   = bc / channels;
    const int tx0 = blockIdx.x * TILE;
    const int ty0 = blockIdx.y * TILE;

    // ---- Gaussian taps: every thread computes the tap sum (15 expf, trivial) ----
    const float s      = sigma[b];
    const float inv2s2 = 1.0f / (2.0f * s * s + 1e-8f);
    float s1 = 0.0f;
#pragma unroll
    for (int k = 0; k < KS; ++k) {
        float d = (float)(k - PADR);
        s1 += expf(-d * d * inv2s2);
    }
    const float invS2 = 1.0f / (s1 * s1 + 1e-8f);   // 2D normalizer = (Σg)^2 + eps

    if (tid < 64) {
        float raw = 0.0f;
        if (tid >= 16 && tid <= 30) {
            float d = (float)(tid - 16 - PADR);
            raw = expf(-d * d * inv2s2);
        }
        hwpad[tid] = raw;           // horizontal pass: raw taps
        vwpad[tid] = raw * invS2;   // vertical pass: carries full normalization
    }

    // ---- stage halo tile: async global->LDS (no VGPR round trip, ASYNCcnt) ----
    const float* xin = x + (size_t)bc * IMG * IMG;
    for (int idx = tid; idx < TROWS * TSTR; idx += 256) {
        int r  = idx / TSTR;
        int c  = idx - r * TSTR;
        int gy = ty0 + r - PADR;
        int gx = tx0 + c - PADR;
        bool inb = (r < IN_T) && (c < IN_T) &&
                   ((unsigned)gy < (unsigned)IMG) && ((unsigned)gx < (unsigned)IMG);
        if (inb) {
            uint32_t ldsoff = (uint32_t)(uintptr_t)(&tile[idx]);
            uint64_t gaddr  = (uint64_t)(uintptr_t)(xin + gy * IMG + gx);
            asm volatile("global_load_async_to_lds_b32 %0, %1, off"
                         :: "v"(ldsoff), "v"(gaddr) : "memory");
        } else {
            tile[idx] = 0.0f;       // zero padding (image border + pad rows/cols)
        }
    }
    asm volatile("s_wait_asynccnt 0" ::: "memory");
    __syncthreads();

    const int lane = tid & 31;
    const int wave = tid >> 5;
    const int m    = lane & 15;     // M (A row) / N (B,D col)
    const int kh   = lane >> 4;     // K-half within 4-wide chunk

    // ---- horizontal pass as banded GEMM: D = X(16x32) * Wh(32x16 banded) ----
    // B chunk t: lanes 0-15 hold k=4t,4t+1 ; lanes 16-31 hold k=4t+2,4t+3
    v2f Bw[8];
#pragma unroll
    for (int t = 0; t < 8; ++t) {
        int j0 = 4 * t + 2 * kh;
        Bw[t].x = hwpad[16 + j0 - m];      // Wh[j][n] = g[j - n]
        Bw[t].y = hwpad[16 + j0 + 1 - m];
    }

#pragma unroll
    for (int rep = 0; rep < 3; ++rep) {    // 20 subtiles over 8 waves
        int st = wave + rep * 8;
        if (st < 20) {                      // wave-uniform guard (EXEC stays all-ones)
            int r0 = (st >> 2) * 16;
            int c0 = (st & 3) * 16;
            v8f acc = {};
#pragma unroll
            for (int t = 0; t < 8; ++t) {
                // A chunk: lane m = row r0+m; cols c0+4t+2kh, +1  (conflict-free b64)
                const float* ap = &tile[(r0 + m) * TSTR + c0 + 4 * t + 2 * kh];
                v2f Av = *(const v2f*)ap;
                acc = __builtin_amdgcn_wmma_f32_16x16x4_f32(
                    false, Av, false, Bw[t], (short)0, acc, false, false);
            }
#pragma unroll
            for (int v = 0; v < 8; ++v)    // D: VGPR v -> row v / v+8 per lane half
                hbuf[(r0 + v + kh * 8) * HSTR + c0 + m] = acc[v];
        }
    }
    __syncthreads();

    // ---- vertical pass as banded GEMM: D = Wv(16x32 banded) * H(32x16) ----
    v2f Amat[8];
#pragma unroll
    for (int t = 0; t < 8; ++t) {
        int j0 = 4 * t + 2 * kh;
        Amat[t].x = vwpad[16 + j0 - m];    // Wv[r][j] = w_v[j - r]
        Amat[t].y = vwpad[16 + j0 + 1 - m];
    }

    float* outp = out + (size_t)bc * IMG * IMG;

#pragma unroll
    for (int i = 0; i < 2; ++i) {          // 16 subtiles over 8 waves
        int st = wave * 2 + i;
        int r0 = (st >> 2) * 16;
        int c0 = (st & 3) * 16;

        v8f acc = {};
#pragma unroll
        for (int t = 0; t < 8; ++t) {
            int row = r0 + 4 * t + 2 * kh; // B chunk: hbuf rows (K), cols c0+m (N)
            v2f Bv;
            Bv.x = hbuf[row * HSTR + c0 + m];
            Bv.y = hbuf[(row + 1) * HSTR + c0 + m];
            acc = __builtin_amdgcn_wmma_f32_16x16x4_f32(
                false, Amat[t], false, Bv, (short)0, acc, false, false);
        }

#pragma unroll
        for (int v = 0; v < 8; ++v) {
            int orow = ty0 + r0 + v + kh * 8;
            int ocol = tx0 + c0 + m;
            outp[orow * IMG + ocol] = acc[v];
        }
    }
}

extern "C" void kernel_launch(void* const* d_in, const int* in_sizes, int n_in,
                              void* d_out, int out_size, void* d_ws, size_t ws_size,
                              hipStream_t stream) {
    const float* x     = (const float*)d_in[0];
    const float* sigma = (const float*)d_in[1];
    float* out         = (float*)d_out;

    const int batch    = in_sizes[1];                 // 32
    const int planes   = in_sizes[0] / (IMG * IMG);   // B*C = 96
    const int channels = planes / batch;              // 3

    dim3 grid(IMG / TILE, IMG / TILE, planes);
    gauss_blur_fused<<<grid, 256, 0, stream>>>(x, sigma, out, channels);
}